// MetaLoss_10642928959862
// MI455X (gfx1250) — compile-verified
//
#include <hip/hip_runtime.h>
#include <hip/hip_bf16.h>

typedef float v2f __attribute__((ext_vector_type(2)));
typedef float v8f __attribute__((ext_vector_type(8)));

#define BB 2048
#define LL 8192
#define GG 256
#define BETA 0.01f

// ---------------------------------------------------------------- utilities
__device__ __forceinline__ float log_sigmoid_neg(float x) {
    // log(sigmoid(-x)) = -softplus(x) = -(max(x,0) + log1p(exp(-|x|)))
    return -(fmaxf(x, 0.0f) + log1pf(__expf(-fabsf(x))));
}

// ---------------------------------------------------------------- kernel 0
__global__ void ml_zero_kernel(float* grp_log, float* acc, int n) {
    int i = blockIdx.x * blockDim.x + threadIdx.x;
    for (; i < n; i += gridDim.x * blockDim.x) grp_log[i] = 0.0f;
    if (blockIdx.x == 0 && threadIdx.x == 0) acc[0] = 0.0f;
}

// ---------------------------------------------------------------- kernel 1
// meta_y[b,g] = 1 iff any true label of group g in row b.
// One block per row; idempotent LDS flag stores (races write identical 1s).
__global__ void ml_metay_kernel(const int* __restrict__ true_y,
                                const int* __restrict__ gid,
                                int* __restrict__ meta) {
    __shared__ int flags[GG];
    const int b = blockIdx.x;
    const int tid = threadIdx.x;
    flags[tid] = 0;
    __syncthreads();
    const size_t base = (size_t)b * LL;
    for (int l = tid; l < LL; l += GG) {
        if (true_y[base + l] != 0) flags[gid[l]] = 1;
    }
    __syncthreads();
    meta[(size_t)b * GG + tid] = flags[tid];
}

// ---------------------------------------------------------------- kernel 2
// Segment sum as GEMM:  grp_log[16 rows x 256 groups] += log_not x OneHot.
// Block: 256 threads = 8 waves.  Wave w handles label chunk
// (blockIdx.y*8 + w)*512 .. +512 for the 16 rows blockIdx.x*16 .. +16.
// Each wave keeps the full 16x256 f32 accumulator tile as 16 v8f regs.
__global__ void __launch_bounds__(256)
ml_segsum_wmma_kernel(const float* __restrict__ logits,
                      const int* __restrict__ gid,
                      float* __restrict__ grp_log) {
    __shared__ float accs[16 * GG];          // 16 KB row-tile accumulator

    const int tid    = threadIdx.x;
    const int lane   = tid & 31;
    const int wave   = tid >> 5;
    const int rowBase = blockIdx.x * 16;
    const int chunk   = blockIdx.y * 8 + wave;      // 0..15
    const int l0Base  = chunk * 512;

    // LDS init
    #pragma unroll
    for (int i = tid; i < 16 * GG; i += 256) accs[i] = 0.0f;
    __syncthreads();

    // Per-lane WMMA element mapping
    const int m   = lane & 15;                // A row / B col
    const int kp  = (lane >> 4) << 1;         // this lane holds K = kp, kp+1

    v8f c[16];
    #pragma unroll
    for (int nt = 0; nt < 16; ++nt) c[nt] = (v8f){0,0,0,0,0,0,0,0};

    const float* rowPtr = logits + (size_t)(rowBase + m) * LL;

    for (int kk = 0; kk < 512; kk += 4) {
        const int l0 = l0Base + kk;
        // A tile: log_not for (row m, K = kp..kp+1)
        v2f a;
        a.x = log_sigmoid_neg(rowPtr[l0 + kp]);
        a.y = log_sigmoid_neg(rowPtr[l0 + kp + 1]);
        // group ids of this lane's two K labels
        const int g0 = gid[l0 + kp];
        const int g1 = gid[l0 + kp + 1];
        // sweep all 16 group tiles (one-hot B built by compare)
        #pragma unroll
        for (int nt = 0; nt < 16; ++nt) {
            const int gBase = nt << 4;
            v2f bv;
            bv.x = (g0 == gBase + m) ? 1.0f : 0.0f;
            bv.y = (g1 == gBase + m) ? 1.0f : 0.0f;
            c[nt] = __builtin_amdgcn_wmma_f32_16x16x4_f32(
                false, a, false, bv, (short)0, c[nt], false, false);
        }
    }

    // C/D layout: VGPR i, lanes0-15 -> M=i ; lanes16-31 -> M=i+8 ; N=lane&15
    const int rHi = (lane >> 4) << 3;
    #pragma unroll
    for (int nt = 0; nt < 16; ++nt) {
        const int grp = (nt << 4) + m;
        #pragma unroll
        for (int i = 0; i < 8; ++i) {
            atomicAdd(&accs[(i + rHi) * GG + grp], c[nt][i]);   // ds_add_f32
        }
    }
    __syncthreads();

    // one global atomic pass per block (two blocks contribute per row tile)
    #pragma unroll
    for (int i = tid; i < 16 * GG; i += 256) {
        const int r = i >> 8, g = i & (GG - 1);
        unsafeAtomicAdd(&grp_log[(size_t)(rowBase + r) * GG + g], accs[i]);
    }
}

// ---------------------------------------------------------------- kernel 3
__global__ void ml_bce_kernel(const float* __restrict__ grp_log,
                              const int* __restrict__ meta,
                              float* __restrict__ acc, int n) {
    __shared__ float red[256];
    float s = 0.0f;
    for (int i = blockIdx.x * blockDim.x + threadIdx.x; i < n;
         i += gridDim.x * blockDim.x) {
        const float gl = grp_log[i];
        const float log_p   = fmaxf(gl, -100.0f);
        const float log_1mp = fmaxf(log1pf(-__expf(gl)), -100.0f);
        s += meta[i] ? log_p : log_1mp;
    }
    red[threadIdx.x] = s;
    __syncthreads();
    for (int off = 128; off > 0; off >>= 1) {
        if (threadIdx.x < off) red[threadIdx.x] += red[threadIdx.x + off];
        __syncthreads();
    }
    if (threadIdx.x == 0) unsafeAtomicAdd(acc, red[0]);
}

// ---------------------------------------------------------------- kernel 4
__global__ void ml_final_kernel(const float* __restrict__ acc, float* out) {
    out[0] = -BETA * acc[0] / (float)(BB * GG);
}

// ---------------------------------------------------------------- launcher
extern "C" void kernel_launch(void* const* d_in, const int* in_sizes, int n_in,
                              void* d_out, int out_size, void* d_ws, size_t ws_size,
                              hipStream_t stream) {
    const float* logits = (const float*)d_in[0];
    const int*   true_y = (const int*)d_in[1];
    const int*   gid    = (const int*)d_in[2];
    float* out = (float*)d_out;

    // workspace layout: grp_log [B*G f32] | meta [B*G i32] | acc [1 f32]
    float* grp_log = (float*)d_ws;
    int*   meta    = (int*)(grp_log + (size_t)BB * GG);
    float* acc     = (float*)(meta + (size_t)BB * GG);

    const int n = BB * GG;

    ml_zero_kernel<<<512, 256, 0, stream>>>(grp_log, acc, n);
    ml_metay_kernel<<<BB, GG, 0, stream>>>(true_y, gid, meta);
    ml_segsum_wmma_kernel<<<dim3(BB / 16, 2), 256, 0, stream>>>(logits, gid, grp_log);
    ml_bce_kernel<<<512, 256, 0, stream>>>(grp_log, meta, acc, n);
    ml_final_kernel<<<1, 1, 0, stream>>>(acc, out);
}